// WeightedSum_20014547599912
// MI455X (gfx1250) — compile-verified
//
#include <hip/hip_runtime.h>
#include <math.h>

typedef float v2f __attribute__((ext_vector_type(2)));
typedef float v4f __attribute__((ext_vector_type(4)));
typedef float v8f __attribute__((ext_vector_type(8)));

#define Bn 128
#define Hn 56
#define Wn 56
#define Cn 512
#define HW (Hn * Wn)                 // 3136
#define TILES_PER_IMG (HW / 16)      // 196 (exact)
#define WAVES_PER_BLOCK 8
#define TOTAL_TILES (Bn * TILES_PER_IMG)  // 25088 = 3136 blocks * 8 waves (exact)

// Kernel 1: batched matvec via V_WMMA_F32_16X16X4_F32.
// One wave computes 16 pixel dot-products (K=512) for one image.
// One global_load_b128 per lane feeds TWO wmma ops: WMMA K-slot assignment is
// arbitrary as long as A and B agree per slot (B is the replicated a-vector),
// so lanes 0-15 carry channels k..k+3 and lanes 16-31 carry k+4..k+7.
__global__ __launch_bounds__(256)
void wsum_wmma_kernel(const float* __restrict__ batch,
                      const float* __restrict__ avec,
                      float* __restrict__ img) {
    const int wave = blockIdx.x * WAVES_PER_BLOCK + (threadIdx.x >> 5);
    const int lane = threadIdx.x & 31;

    const int b = wave / TILES_PER_IMG;
    const int t = wave % TILES_PER_IMG;

    const int m    = lane & 15;          // matrix row (pixel within tile)
    const int koff = (lane >> 4) << 2;   // channel sub-offset per lane group: 0 or 4

    // Lane's pixel-row base (rows are contiguous 512 f32 = 2KB in BHWC layout)
    const float* __restrict__ pA =
        batch + ((size_t)(b * HW + t * 16 + m) * Cn) + koff;
    // B operand: a_batch[b] replicated across all 16 N columns -> per-lane
    // value is a[b, same channels as this lane's A slots] (WGP$/L2-resident).
    const float* __restrict__ pB = avec + (size_t)b * Cn + koff;

    v8f acc = {0.f, 0.f, 0.f, 0.f, 0.f, 0.f, 0.f, 0.f};

    for (int kk = 0; kk < Cn; kk += 64) {
        if (kk + 64 < Cn) {
            __builtin_prefetch(pA + kk + 64, 0, 1);   // global_prefetch_b8, 256B ahead
        }
#pragma unroll
        for (int k8 = 0; k8 < 64; k8 += 8) {
            v4f a4 = *(const v4f*)(pA + kk + k8);     // global_load_b128
            v4f b4 = *(const v4f*)(pB + kk + k8);     // global_load_b128 (cache-hit)
            v2f aLo = {a4.x, a4.y};
            v2f bLo = {b4.x, b4.y};
            v2f aHi = {a4.z, a4.w};
            v2f bHi = {b4.z, b4.w};
            // 8 args: (neg_a, A, neg_b, B, c_mod, C, reuse_a, reuse_b)
            acc = __builtin_amdgcn_wmma_f32_16x16x4_f32(
                false, aLo, false, bLo, (short)0, acc, false, false);
            acc = __builtin_amdgcn_wmma_f32_16x16x4_f32(
                false, aHi, false, bHi, (short)0, acc, false, false);
        }
    }

    // D layout: lanes 0-15 hold M=0..7 in acc[0..7]; lanes 16-31 hold M=8..15.
    // All 16 columns are identical, so lane 0 and lane 16 carry the results.
    if ((lane & 15) == 0) {
        float* dst = img + (size_t)b * HW + (size_t)t * 16 + ((lane >> 4) << 3);
        v4f lo = {acc[0], acc[1], acc[2], acc[3]};
        v4f hi = {acc[4], acc[5], acc[6], acc[7]};
        *(v4f*)(dst)     = lo;   // global_store_b128
        *(v4f*)(dst + 4) = hi;
    }
}

// Kernel 2: per-image max reduction + normalize (3.2 MB total traffic, trivial).
__global__ __launch_bounds__(256)
void norm_kernel(const float* __restrict__ img, float* __restrict__ out) {
    __shared__ float red[256];
    const int b   = blockIdx.x;
    const int tid = threadIdx.x;

    const float* __restrict__ src = img + (size_t)b * HW;
    float m = -INFINITY;
    for (int i = tid; i < HW; i += 256) m = fmaxf(m, src[i]);
    red[tid] = m;
    __syncthreads();
    for (int s = 128; s > 0; s >>= 1) {
        if (tid < s) red[tid] = fmaxf(red[tid], red[tid + s]);
        __syncthreads();
    }
    const float mx = red[0];

    float* __restrict__ dst = out + (size_t)b * HW;
    for (int i = tid; i < HW; i += 256) dst[i] = src[i] / mx;
}

extern "C" void kernel_launch(void* const* d_in, const int* in_sizes, int n_in,
                              void* d_out, int out_size, void* d_ws, size_t ws_size,
                              hipStream_t stream) {
    const float* batch = (const float*)d_in[0];   // [B,H,W,C] f32
    const float* avec  = (const float*)d_in[1];   // [B,C]     f32
    float* out = (float*)d_out;                   // [B,H,W]   f32
    float* img = (float*)d_ws;                    // scratch: B*HW floats (~1.6 MB)

    wsum_wmma_kernel<<<TOTAL_TILES / WAVES_PER_BLOCK, 256, 0, stream>>>(batch, avec, img);
    norm_kernel<<<Bn, 256, 0, stream>>>(img, out);
}